// GraphAttention_48198122996114
// MI455X (gfx1250) — compile-verified
//
#include <hip/hip_runtime.h>

#define D_FEAT 128
#define UNITS  64

typedef __attribute__((ext_vector_type(2))) float v2f;
typedef __attribute__((ext_vector_type(8))) float v8f;

#define GEMM_WAVES 8
#define ROWS_PER_BLOCK (GEMM_WAVES * 16)
#define LDSB_PITCH 66   // 64 + 2 pad: avoids LDS bank conflicts on row-strided frag reads

// -------------------- WMMA fp32 GEMM: H[n,64] = A[n,128] x B[128,64] --------------------
__global__ __launch_bounds__(256) void gat_gemm_kernel(const float* __restrict__ A,
                                                       const float* __restrict__ B,
                                                       float* __restrict__ H, int n) {
    __shared__ float ldsB[D_FEAT * LDSB_PITCH];   // 33792 B

    // Stage B (128x64 fp32 = 32KB) into padded LDS, coalesced float4 loads.
    for (int i = threadIdx.x; i < (D_FEAT * UNITS) / 4; i += 256) {
        float4 v = ((const float4*)B)[i];
        int r = i >> 4;              // 16 float4 per 64-wide row
        int c = (i & 15) << 2;
        float* p = &ldsB[r * LDSB_PITCH + c];
        p[0] = v.x; p[1] = v.y; p[2] = v.z; p[3] = v.w;
    }
    __syncthreads();

    const int wave = threadIdx.x >> 5;
    const int lane = threadIdx.x & 31;
    const int half = lane >> 4;      // 0: lanes 0-15, 1: lanes 16-31
    const int l16  = lane & 15;

    const int row0 = blockIdx.x * ROWS_PER_BLOCK + wave * 16;
    int arow = row0 + l16;
    if (arow >= n) arow = n - 1;     // clamp so EXEC stays all-1s for WMMA (stores predicated)

    // A-frag layout (16x4 f32): VGPR v, lane -> K = 2*half + v; M = l16
    const float* aptr = A + (size_t)arow * D_FEAT + 2 * half;

    v8f acc0 = {}, acc1 = {}, acc2 = {}, acc3 = {};

    #pragma unroll 4
    for (int k = 0; k < D_FEAT; k += 4) {
        v2f a;
        a.x = aptr[k + 0];
        a.y = aptr[k + 1];

        // B-frag (4x16 f32): VGPR v, lane -> K = k + 2*half + v; N = 16*j + l16
        const float* b0 = &ldsB[(k + 2 * half) * LDSB_PITCH + l16];
        const float* b1 = b0 + LDSB_PITCH;

        v2f bf;
        bf.x = b0[0];  bf.y = b1[0];
        acc0 = __builtin_amdgcn_wmma_f32_16x16x4_f32(false, a, false, bf, (short)0, acc0, false, false);
        bf.x = b0[16]; bf.y = b1[16];
        acc1 = __builtin_amdgcn_wmma_f32_16x16x4_f32(false, a, false, bf, (short)0, acc1, false, false);
        bf.x = b0[32]; bf.y = b1[32];
        acc2 = __builtin_amdgcn_wmma_f32_16x16x4_f32(false, a, false, bf, (short)0, acc2, false, false);
        bf.x = b0[48]; bf.y = b1[48];
        acc3 = __builtin_amdgcn_wmma_f32_16x16x4_f32(false, a, false, bf, (short)0, acc3, false, false);
    }

    // C/D layout: VGPR r -> M = row0 + r + 8*half, N = 16*j + l16
    #pragma unroll
    for (int r = 0; r < 8; ++r) {
        int row = row0 + r + 8 * half;
        if (row < n) {
            float* hp = H + (size_t)row * UNITS + l16;
            hp[0]  = acc0[r];
            hp[16] = acc1[r];
            hp[32] = acc2[r];
            hp[48] = acc3[r];
        }
    }
}

// -------------------- per-node attention projections --------------------
__global__ __launch_bounds__(256) void gat_proj_kernel(const float* __restrict__ H,
                                                       const float* __restrict__ ka,  // [2*UNITS]
                                                       float* __restrict__ s_src,
                                                       float* __restrict__ s_dst, int n) {
    int i = blockIdx.x * 256 + threadIdx.x;
    if (i >= n) return;
    const float4* h4  = (const float4*)(H + (size_t)i * UNITS);
    const float4* as4 = (const float4*)ka;          // a_src = ka[0:64]
    const float4* ad4 = (const float4*)(ka + UNITS); // a_dst = ka[64:128]
    float ss = 0.f, sd = 0.f;
    #pragma unroll
    for (int c = 0; c < UNITS / 4; ++c) {
        float4 v = h4[c];
        float4 s = as4[c];
        float4 d = ad4[c];
        ss += v.x * s.x + v.y * s.y + v.z * s.z + v.w * s.w;
        sd += v.x * d.x + v.y * d.y + v.z * d.z + v.w * d.w;
    }
    s_src[i] = ss;
    s_dst[i] = sd;
}

// -------------------- per-edge scores + denom segment-sum --------------------
__global__ __launch_bounds__(256) void gat_score_kernel(const int* __restrict__ edges,
                                                        const float* __restrict__ s_src,
                                                        const float* __restrict__ s_dst,
                                                        float* __restrict__ scores,
                                                        float* __restrict__ denom, int ne) {
    int e = blockIdx.x * 256 + threadIdx.x;
    if (e >= ne) return;
    int s = edges[2 * e];
    int d = edges[2 * e + 1];
    float sc = s_src[s] + s_dst[d];
    sc = (sc > 0.f) ? sc : 0.2f * sc;          // leaky_relu(0.2)
    sc = fminf(2.0f, fmaxf(-2.0f, sc));        // clip
    sc = __expf(sc);
    scores[e] = sc;
    atomicAdd(denom + s, sc);
}

// -------------------- per-edge weighted aggregation: 16 lanes per edge --------------------
__global__ __launch_bounds__(256) void gat_agg_kernel(const int* __restrict__ edges,
                                                      const float* __restrict__ scores,
                                                      const float* __restrict__ denom,
                                                      const float* __restrict__ H,
                                                      float* __restrict__ out, int ne) {
    long long gid = (long long)blockIdx.x * 256 + threadIdx.x;
    int e = (int)(gid >> 4);
    int l = (int)(gid & 15);
    if (e >= ne) return;
    int s = edges[2 * e];
    int d = edges[2 * e + 1];
    float alpha = scores[e] / denom[s];
    float4 hv = ((const float4*)(H + (size_t)d * UNITS))[l];
    float* op = out + (size_t)s * UNITS + 4 * l;
    atomicAdd(op + 0, alpha * hv.x);
    atomicAdd(op + 1, alpha * hv.y);
    atomicAdd(op + 2, alpha * hv.z);
    atomicAdd(op + 3, alpha * hv.w);
}

extern "C" void kernel_launch(void* const* d_in, const int* in_sizes, int n_in,
                              void* d_out, int out_size, void* d_ws, size_t ws_size,
                              hipStream_t stream) {
    const float* node_states = (const float*)d_in[0];
    const int*   edges       = (const int*)d_in[1];     // [ne, 2] int32
    const float* kernelW     = (const float*)d_in[2];   // [128, 64]
    const float* ka          = (const float*)d_in[3];   // [128, 1]
    float* out = (float*)d_out;

    const int n  = in_sizes[0] / D_FEAT;
    const int ne = in_sizes[1] / 2;

    // workspace layout (floats): H[n*64] | s_src[n] | s_dst[n] | denom[n] | scores[ne]
    float* H      = (float*)d_ws;
    float* s_src  = H + (size_t)n * UNITS;
    float* s_dst  = s_src + n;
    float* denom  = s_dst + n;
    float* scores = denom + n;

    hipMemsetAsync(out,   0, (size_t)n * UNITS * sizeof(float), stream);
    hipMemsetAsync(denom, 0, (size_t)n * sizeof(float), stream);

    gat_gemm_kernel<<<(n + ROWS_PER_BLOCK - 1) / ROWS_PER_BLOCK, 256, 0, stream>>>(
        node_states, kernelW, H, n);

    gat_proj_kernel<<<(n + 255) / 256, 256, 0, stream>>>(H, ka, s_src, s_dst, n);

    gat_score_kernel<<<(ne + 255) / 256, 256, 0, stream>>>(edges, s_src, s_dst, scores, denom, ne);

    long long agg_threads = (long long)ne * 16;
    gat_agg_kernel<<<(unsigned)((agg_threads + 255) / 256), 256, 0, stream>>>(
        edges, scores, denom, H, out, ne);
}